// MeanCovar_90924457656745
// MI455X (gfx1250) — compile-verified
//
#include <hip/hip_runtime.h>
#include <math.h>

#define B_N 16384
#define C_N 128
#define D_N 64

typedef __attribute__((ext_vector_type(2))) float v2f;
typedef __attribute__((ext_vector_type(8))) float v8f;

// ---------------- zero scratch ----------------
__global__ void k_zero(float* __restrict__ total, unsigned* __restrict__ cnt,
                       unsigned* __restrict__ offs, unsigned* __restrict__ cursor,
                       float* __restrict__ p) {
  int t = blockIdx.x * blockDim.x + threadIdx.x;
  if (t < C_N * D_N) total[t] = 0.f;
  if (t < C_N) { cnt[t] = 0u; offs[t] = 0u; cursor[t] = 0u; }
  if (t < 2)   p[t] = 0.f;
}

// ---------------- argmax + diff + segment count/total ----------------
// one wave32 per sample
__global__ void k_assign(const float* __restrict__ emb, const float* __restrict__ cen,
                         const float* __restrict__ logits, float* __restrict__ diff,
                         int* __restrict__ assign, unsigned* __restrict__ cnt,
                         float* __restrict__ total) {
  int wave = (blockIdx.x * blockDim.x + threadIdx.x) >> 5;
  int lane = threadIdx.x & 31;
  if (wave >= B_N) return;
  const float* lrow = logits + (size_t)wave * C_N;
  float best = -3.0e38f; int bi = 0;
#pragma unroll
  for (int j = 0; j < 4; ++j) {
    int c = lane + 32 * j;
    float v = lrow[c];
    if (v > best) { best = v; bi = c; }          // first-max within lane
  }
#pragma unroll
  for (int off = 16; off >= 1; off >>= 1) {      // wave32 butterfly argmax
    float ov = __shfl_xor(best, off, 32);
    int   oi = __shfl_xor(bi,   off, 32);
    if (ov > best || (ov == best && oi < bi)) { best = ov; bi = oi; }
  }
  const float* erow = emb + (size_t)wave * D_N;
  const float* crow = cen + (size_t)bi * D_N;
  float d0 = erow[lane]      - crow[lane];
  float d1 = erow[lane + 32] - crow[lane + 32];
  diff[(size_t)wave * D_N + lane]      = d0;
  diff[(size_t)wave * D_N + lane + 32] = d1;
  atomicAdd(&total[bi * D_N + lane],      d0);
  atomicAdd(&total[bi * D_N + lane + 32], d1);
  if (lane == 0) { assign[wave] = bi; atomicAdd(&cnt[bi], 1u); }
}

// ---------------- CSR offsets (C=128, trivial serial scan) ----------------
__global__ void k_scan(const unsigned* __restrict__ cnt, unsigned* __restrict__ offs,
                       unsigned* __restrict__ cursor) {
  if (threadIdx.x == 0 && blockIdx.x == 0) {
    unsigned run = 0;
    for (int c = 0; c < C_N; ++c) { offs[c] = run; cursor[c] = run; run += cnt[c]; }
  }
}

// ---------------- CSR fill ----------------
__global__ void k_fill(const int* __restrict__ assign, unsigned* __restrict__ cursor,
                       int* __restrict__ bucket) {
  int b = blockIdx.x * blockDim.x + threadIdx.x;
  if (b >= B_N) return;
  int c = assign[b];
  unsigned pos = atomicAdd(&cursor[c], 1u);
  bucket[pos] = b;
}

// ---------------- moment1 penalty ----------------
__global__ void k_pen1(const float* __restrict__ total, const unsigned* __restrict__ cnt,
                       float* __restrict__ p) {
  int t = blockIdx.x * blockDim.x + threadIdx.x;
  float local = 0.f;
  if (t < C_N * D_N) {
    int c = t / D_N;
    float n  = (float)cnt[c];
    float m1 = total[t] / (n + 1e-8f);
    float w  = (n / (float)B_N) * (1.0f / (float)D_N);
    local = w * m1 * m1;
  }
#pragma unroll
  for (int off = 16; off >= 1; off >>= 1) local += __shfl_xor(local, off, 32);
  if ((threadIdx.x & 31) == 0) atomicAdd(&p[0], local);
}

// ---------------- per-cluster Gram matrix via V_WMMA_F32_16X16X4_F32 ----------------
// One block (16 waves) per cluster; wave w owns the 16x16 tile (ti=w/4, tj=w%4)
// of the 64x64 moment2. LDS stages 32 gathered diff rows per chunk.
__global__ void __launch_bounds__(512) k_m2(const float* __restrict__ diff,
                                            const int* __restrict__ bucket,
                                            const unsigned* __restrict__ offs,
                                            const unsigned* __restrict__ cnt,
                                            float* __restrict__ p) {
  __shared__ float lds[32 * D_N];   // 8 KB of 320 KB/WGP
  int c = blockIdx.x;
  unsigned n    = cnt[c];
  unsigned base = offs[c];
  int wave = threadIdx.x >> 5;      // 0..15
  int lane = threadIdx.x & 31;
  int ti = wave >> 2, tj = wave & 3;
  int half = lane >> 4;             // 0: K lo pair, 1: K hi pair
  int l    = lane & 15;

  v8f acc = {};
  int nchunks = (int)((n + 31u) >> 5);
  for (int ch = 0; ch < nchunks; ++ch) {
    __syncthreads();                // protect LDS from previous iteration readers
    int r  = threadIdx.x >> 4;      // 0..31 staged row
    int cs = (threadIdx.x & 15) << 2;
    int si = ch * 32 + r;
    float4 v = make_float4(0.f, 0.f, 0.f, 0.f);   // zero-pad remainder rows
    if ((unsigned)si < n) {
      int s = bucket[base + si];
      v = *(const float4*)&diff[(size_t)s * D_N + cs];
    }
    *(float4*)&lds[r * D_N + cs] = v;
    __syncthreads();
#pragma unroll
    for (int kk = 0; kk < 8; ++kk) {              // 8 K-steps of 4 samples
      int k0 = kk * 4;
      v2f a, b;
      // fp32 A 16x4 layout: lanes 0-15 -> (K=0,K=1), lanes 16-31 -> (K=2,K=3); M = lane%16
      a.x = lds[(k0 + 2 * half + 0) * D_N + ti * 16 + l];
      a.y = lds[(k0 + 2 * half + 1) * D_N + ti * 16 + l];
      // fp32 B 4x16 mirrored layout: N = lane%16
      b.x = lds[(k0 + 2 * half + 0) * D_N + tj * 16 + l];
      b.y = lds[(k0 + 2 * half + 1) * D_N + tj * 16 + l];
      acc = __builtin_amdgcn_wmma_f32_16x16x4_f32(false, a, false, b,
                                                  (short)0, acc, false, false);
    }
  }

  // fused normalize + signed-sqrt compression + weighted penalty2 partial
  float fn  = (float)n;
  float inv = 1.f / (fn + 1e-8f);
  float cwn = fn / (float)B_N;
  const float Aw = 1.f / (2.f * (float)D_N);
  const float Bw = 1.f / (2.f * (float)D_N * (float)(D_N - 1));
  float local = 0.f;
#pragma unroll
  for (int r2 = 0; r2 < 8; ++r2) {
    // C/D 16x16 f32 layout: VGPR r2 -> M = r2 + 8*half, N = lane%16
    int gi = ti * 16 + r2 + 8 * half;
    int gj = tj * 16 + l;
    float m  = acc[r2] * inv;
    float s  = (m < 0.f) ? -1.f : 1.f;
    float cm = s * (sqrtf(fabsf(m) + 0.25f) - 0.5f);
    float e  = (gi == gj) ? 1.f : 0.f;
    float t  = cm - e;
    float w  = cwn * ((gi == gj) ? Aw : Bw);
    local += w * t * t;
  }
#pragma unroll
  for (int off = 16; off >= 1; off >>= 1) local += __shfl_xor(local, off, 32);
  if (lane == 0) atomicAdd(&p[1], local);
}

// ---------------- finalize ----------------
__global__ void k_final(const float* __restrict__ p, float* __restrict__ out) {
  if (threadIdx.x == 0 && blockIdx.x == 0) out[0] = p[0] + 0.5f * p[1];
}

extern "C" void kernel_launch(void* const* d_in, const int* in_sizes, int n_in,
                              void* d_out, int out_size, void* d_ws, size_t ws_size,
                              hipStream_t stream) {
  const float* emb    = (const float*)d_in[0];   // [B, D]
  const float* cen    = (const float*)d_in[1];   // [C, D]
  const float* logits = (const float*)d_in[2];   // [B, C]
  float* out = (float*)d_out;
  (void)in_sizes; (void)n_in; (void)out_size; (void)ws_size;

  char* w = (char*)d_ws;
  float*    diff   = (float*)w;    w += (size_t)B_N * D_N * sizeof(float);  // 4 MB
  int*      assign = (int*)w;      w += (size_t)B_N * sizeof(int);
  int*      bucket = (int*)w;      w += (size_t)B_N * sizeof(int);
  unsigned* cnt    = (unsigned*)w; w += C_N * sizeof(unsigned);
  unsigned* offs   = (unsigned*)w; w += C_N * sizeof(unsigned);
  unsigned* cursor = (unsigned*)w; w += C_N * sizeof(unsigned);
  float*    total  = (float*)w;    w += (size_t)C_N * D_N * sizeof(float);
  float*    p      = (float*)w;    w += 2 * sizeof(float);

  k_zero  <<<(C_N * D_N + 255) / 256, 256, 0, stream>>>(total, cnt, offs, cursor, p);
  k_assign<<<(B_N * 32) / 256, 256, 0, stream>>>(emb, cen, logits, diff, assign, cnt, total);
  k_scan  <<<1, 32, 0, stream>>>(cnt, offs, cursor);
  k_fill  <<<B_N / 256, 256, 0, stream>>>(assign, cursor, bucket);
  k_pen1  <<<(C_N * D_N + 255) / 256, 256, 0, stream>>>(total, cnt, p);
  k_m2    <<<C_N, 512, 0, stream>>>(diff, bucket, offs, cnt, p);
  k_final <<<1, 32, 0, stream>>>(p, out);
}